// JavaEncoder_10075993276846
// MI455X (gfx1250) — compile-verified
//
#include <hip/hip_runtime.h>
#include <hip/hip_bf16.h>

#define N_NODES   100000
#define N_EDGES   1600000
#define NFEAT     128
#define DIM       64
#define NUM_GRAPHS 64

typedef __attribute__((ext_vector_type(2))) float v2f;
typedef __attribute__((ext_vector_type(8))) float v8f;

// ---------------------------------------------------------------------------
// zero fill
// ---------------------------------------------------------------------------
__global__ void zero_kernel(float* __restrict__ p, long long n) {
    long long i = (long long)blockIdx.x * blockDim.x + threadIdx.x;
    long long stride = (long long)gridDim.x * blockDim.x;
    for (; i < n; i += stride) p[i] = 0.0f;
}

// ---------------------------------------------------------------------------
// out = relu(x @ w_lin0 + b_lin0)   [N,128] x [128,64] -> [N,64]
// One wave computes one 16x16 tile using V_WMMA_F32_16X16X4_F32 (K-step 4).
// ---------------------------------------------------------------------------
__global__ void lin0_kernel(const float* __restrict__ x,
                            const float* __restrict__ w,
                            const float* __restrict__ bias,
                            float* __restrict__ out) {
    const int waves_per_block = blockDim.x >> 5;
    const int wave = blockIdx.x * waves_per_block + (threadIdx.x >> 5);
    const int lane = threadIdx.x & 31;
    const int tiles_n = DIM / 16;                 // 4
    const int m0 = (wave / tiles_n) * 16;
    const int n0 = (wave % tiles_n) * 16;
    if (m0 >= N_NODES) return;                    // wave-uniform; EXEC stays full

    const int l16 = lane & 15;
    const int hi  = lane >> 4;                    // 0: K 0..1 / M 0..7, 1: K 2..3 / M 8..15

    v8f c = {};
    const float* arow = x + (long long)(m0 + l16) * NFEAT;
    for (int k0 = 0; k0 < NFEAT; k0 += 4) {
        const int ka = k0 + 2 * hi;               // even -> 8B aligned
        v2f a = *(const v2f*)(arow + ka);
        v2f b;
        b.x = w[(ka    ) * DIM + n0 + l16];
        b.y = w[(ka + 1) * DIM + n0 + l16];
        c = __builtin_amdgcn_wmma_f32_16x16x4_f32(false, a, false, b,
                                                  (short)0, c, false, false);
    }

    const float bv = bias[n0 + l16];
#pragma unroll
    for (int r = 0; r < 8; ++r) {
        const int m = m0 + r + 8 * hi;
        out[(long long)m * DIM + n0 + l16] = fmaxf(c[r] + bv, 0.0f);
    }
}

// ---------------------------------------------------------------------------
// deg[src[e]] += 1
// ---------------------------------------------------------------------------
__global__ void deg_kernel(const int* __restrict__ src, float* __restrict__ deg) {
    int e = blockIdx.x * blockDim.x + threadIdx.x;
    if (e < N_EDGES) atomicAdd(&deg[src[e]], 1.0f);
}

// ---------------------------------------------------------------------------
// dinv[i] = deg>0 ? rsqrt(deg) : 0
// ---------------------------------------------------------------------------
__global__ void dinv_kernel(const float* __restrict__ deg, float* __restrict__ dinv) {
    int i = blockIdx.x * blockDim.x + threadIdx.x;
    if (i < N_NODES) {
        float d = deg[i];
        dinv[i] = (d > 0.0f) ? (1.0f / sqrtf(d)) : 0.0f;
    }
}

// ---------------------------------------------------------------------------
// tx1[dst] += (-dinv[src]*dinv[dst]) * out[src]
// 16 lanes per edge, 4 consecutive floats per lane (float4 gather).
// ---------------------------------------------------------------------------
__global__ void scatter_kernel(const int* __restrict__ src,
                               const int* __restrict__ dst,
                               const float* __restrict__ dinv,
                               const float* __restrict__ out,
                               float* __restrict__ tx1) {
    long long idx = (long long)blockIdx.x * blockDim.x + threadIdx.x;
    if (idx >= (long long)N_EDGES * 16) return;
    const int e = (int)(idx >> 4);
    const int q = (int)(idx & 15) * 4;
    const int s = src[e];
    const int d = dst[e];
    const float norm = -dinv[s] * dinv[d];
    const float4 v = *(const float4*)(out + (long long)s * DIM + q);
    float* t = tx1 + (long long)d * DIM + q;
    atomicAdd(t + 0, norm * v.x);
    atomicAdd(t + 1, norm * v.y);
    atomicAdd(t + 2, norm * v.z);
    atomicAdd(t + 3, norm * v.w);
}

// ---------------------------------------------------------------------------
// h = relu(tx0 @ w0 + tx1 @ w1 + b_cheb)   two K=64 GEMMs into one accumulator
// ---------------------------------------------------------------------------
__global__ void cheb_kernel(const float* __restrict__ tx0,
                            const float* __restrict__ tx1,
                            const float* __restrict__ w0,
                            const float* __restrict__ w1,
                            const float* __restrict__ bias,
                            float* __restrict__ h) {
    const int waves_per_block = blockDim.x >> 5;
    const int wave = blockIdx.x * waves_per_block + (threadIdx.x >> 5);
    const int lane = threadIdx.x & 31;
    const int tiles_n = DIM / 16;
    const int m0 = (wave / tiles_n) * 16;
    const int n0 = (wave % tiles_n) * 16;
    if (m0 >= N_NODES) return;

    const int l16 = lane & 15;
    const int hi  = lane >> 4;

    v8f c = {};
    const float* a0 = tx0 + (long long)(m0 + l16) * DIM;
    const float* a1 = tx1 + (long long)(m0 + l16) * DIM;
    for (int k0 = 0; k0 < DIM; k0 += 4) {
        const int ka = k0 + 2 * hi;
        v2f a = *(const v2f*)(a0 + ka);
        v2f b;
        b.x = w0[(ka    ) * DIM + n0 + l16];
        b.y = w0[(ka + 1) * DIM + n0 + l16];
        c = __builtin_amdgcn_wmma_f32_16x16x4_f32(false, a, false, b,
                                                  (short)0, c, false, false);
    }
    for (int k0 = 0; k0 < DIM; k0 += 4) {
        const int ka = k0 + 2 * hi;
        v2f a = *(const v2f*)(a1 + ka);
        v2f b;
        b.x = w1[(ka    ) * DIM + n0 + l16];
        b.y = w1[(ka + 1) * DIM + n0 + l16];
        c = __builtin_amdgcn_wmma_f32_16x16x4_f32(false, a, false, b,
                                                  (short)0, c, false, false);
    }

    const float bv = bias[n0 + l16];
#pragma unroll
    for (int r = 0; r < 8; ++r) {
        const int m = m0 + r + 8 * hi;
        h[(long long)m * DIM + n0 + l16] = fmaxf(c[r] + bv, 0.0f);
    }
}

// ---------------------------------------------------------------------------
// pooled[g] = sum over nodes with batch==g of h — LDS accumulator per block
// (64 graphs x 64 dims = 4096 f32 = 16KB of the 320KB WGP LDS)
// ---------------------------------------------------------------------------
__global__ void pool_kernel(const float* __restrict__ h,
                            const int* __restrict__ batch,
                            float* __restrict__ pooled) {
    __shared__ float acc[NUM_GRAPHS * DIM];
    for (int i = threadIdx.x; i < NUM_GRAPHS * DIM; i += blockDim.x) acc[i] = 0.0f;
    __syncthreads();

    const long long total = (long long)N_NODES * DIM;
    const long long stride = (long long)gridDim.x * blockDim.x;
    for (long long idx = (long long)blockIdx.x * blockDim.x + threadIdx.x;
         idx < total; idx += stride) {
        const int node = (int)(idx >> 6);
        const int dd   = (int)(idx & 63);
        const int g    = batch[node];
        atomicAdd(&acc[g * DIM + dd], h[idx]);
    }
    __syncthreads();

    for (int i = threadIdx.x; i < NUM_GRAPHS * DIM; i += blockDim.x)
        atomicAdd(&pooled[i], acc[i]);
}

// ---------------------------------------------------------------------------
extern "C" void kernel_launch(void* const* d_in, const int* in_sizes, int n_in,
                              void* d_out, int out_size, void* d_ws, size_t ws_size,
                              hipStream_t stream) {
    const float* x      = (const float*)d_in[0];
    const int*   eidx   = (const int*)d_in[1];
    const int*   batch  = (const int*)d_in[2];
    const float* w_lin0 = (const float*)d_in[3];
    const float* b_lin0 = (const float*)d_in[4];
    const float* w0     = (const float*)d_in[5];
    const float* w1     = (const float*)d_in[6];
    const float* b_cheb = (const float*)d_in[7];

    float* pooled = (float*)d_out;                     // [64,64] = 4096 floats
    float* h      = pooled + NUM_GRAPHS * DIM;         // [N,64]

    char* ws = (char*)d_ws;
    float* out  = (float*)(ws);                                       // [N,64]
    float* tx1  = (float*)(ws + (size_t)N_NODES * DIM * 4);           // [N,64]
    float* deg  = (float*)(ws + (size_t)N_NODES * DIM * 8);           // [N]
    float* dinv = (float*)(ws + (size_t)N_NODES * DIM * 8 + (size_t)N_NODES * 4);

    const int* src = eidx;
    const int* dst = eidx + N_EDGES;

    // zero accumulators (every call: deterministic under graph replay)
    zero_kernel<<<2048, 256, 0, stream>>>(tx1, (long long)N_NODES * DIM);
    zero_kernel<<<256, 256, 0, stream>>>(deg, (long long)N_NODES);
    zero_kernel<<<16, 256, 0, stream>>>(pooled, (long long)NUM_GRAPHS * DIM);

    // 1) out = relu(x @ w_lin0 + b)
    {
        const int tiles = (N_NODES / 16) * (DIM / 16);   // 25000 waves
        const int wpb = 8;                                // 256 threads
        lin0_kernel<<<tiles / wpb, 32 * wpb, 0, stream>>>(x, w_lin0, b_lin0, out);
    }

    // 2) degree + dinv
    deg_kernel<<<(N_EDGES + 255) / 256, 256, 0, stream>>>(src, deg);
    dinv_kernel<<<(N_NODES + 255) / 256, 256, 0, stream>>>(deg, dinv);

    // 3) edge scatter: tx1[dst] += norm * out[src]
    {
        const long long threads = (long long)N_EDGES * 16;
        scatter_kernel<<<(unsigned)((threads + 255) / 256), 256, 0, stream>>>(
            src, dst, dinv, out, tx1);
    }

    // 4) h = relu(out @ w0 + tx1 @ w1 + b_cheb)
    {
        const int tiles = (N_NODES / 16) * (DIM / 16);
        const int wpb = 8;
        cheb_kernel<<<tiles / wpb, 32 * wpb, 0, stream>>>(out, tx1, w0, w1, b_cheb, h);
    }

    // 5) pooled = segment_sum(h, batch)
    pool_kernel<<<512, 256, 0, stream>>>(h, batch, pooled);
}